// SampleQueryExtractionLayer_86852828660039
// MI455X (gfx1250) — compile-verified
//
#include <hip/hip_runtime.h>

// SampleQueryExtraction on MI455X (gfx1250, wave32) — round 2.
//
// mask = relu(1+eps - L1(query, grid))^2 is 4-sparse per query (4 integer
// corners). Output row = sum_j w_j * features[idx_j]. Exact-f32 version:
//
//  * Gather: 64 rows x 128 ch (f32) DMA'd into LDS with
//    global_load_async_to_lds_b128 (ASYNCcnt path, no VGPR staging).
//  * Compute: D(16x16) += A_t(16x4) x B_t(4x16) via 16 chained
//    v_wmma_f32_16x16x4_f32; WMMA t carries query t's 4 corner weights
//    (A nonzero only in row t -> 2 v_cndmask, no LDS for A).
//  * Roofline: ~26 MB HBM traffic -> ~1.1 us @ 23.3 TB/s; FLOPs are free
//    (0.27 GFLOP effective). Features (16.8 MB) are L2-resident (192 MB).

typedef __attribute__((ext_vector_type(2))) float v2f;
typedef __attribute__((ext_vector_type(8))) float v8f;

#define B_    4
#define S_    8
#define Q_    256
#define C_    256
#define N_    4096
#define SIZE_ 64
#define EPS_  1.0e-4f

#define CH_   128          // channels per workgroup (C_/2)
#define STR_  136          // LDS row stride in floats (128 + 8 pad -> no bank conflicts)

__global__ __launch_bounds__(256)
void sqe_wmma_f32_kernel(const float* __restrict__ features,
                         const float* __restrict__ qp,
                         float* __restrict__ out) {
  __shared__ __align__(16) float ldsF[64 * STR_];   // [slot k][channel], ~34.8 KB

  const int tid  = threadIdx.x;
  const int wg   = blockIdx.x;          // 1024 = B*S*(Q/16) * 2 channel halves
  const int half = wg & 1;              // which 128-channel half
  const int tile = wg >> 1;
  const int qt   = tile & 15;
  const int s    = (tile >> 4) & 7;
  const int b    = tile >> 7;
  const int q0   = qt * 16;

  const int lane = tid & 31;
  const int wave = tid >> 5;            // 8 waves -> 8 channel blocks of 16
  const int row  = lane & 15;           // M index this lane owns in A; N index in B/D
  const int hi   = lane >> 4;           // lane half selects K sub-range

  // ---- Per-lane corner weights for query (q0+row): exact f32 ----
  float wA, wB;                          // normalized weights for K = hi*2, hi*2+1
  {
    const size_t qb = ((size_t)(b * S_ + s) * Q_ + q0 + row) * 2;
    const float py = qp[qb + 0];
    const float px = qp[qb + 1];
    const float fy = py - floorf(py);
    const float fx = px - floorf(px);
    float w[4];
#pragma unroll
    for (int j = 0; j < 4; ++j) {
      const float dy = (j >> 1) ? (1.0f - fy) : fy;
      const float dx = (j & 1)  ? (1.0f - fx) : fx;
      float t = 1.0f + EPS_ - dy - dx;
      t = t > 0.0f ? t : 0.0f;
      w[j] = t * t;                      // GAMMA = 2
    }
    const float inv = 1.0f / (w[0] + w[1] + w[2] + w[3] + EPS_);
    wA = w[hi * 2 + 0] * inv;
    wB = w[hi * 2 + 1] * inv;
  }

  // ---- Async DMA gather: 64 rows x 128 ch f32 -> LDS, one row per slot k, channels along the row ----
  // 16 threads per row; lane l copies 16B chunks c4 = l and l+16.
  {
    const unsigned lds_base = (unsigned)(uintptr_t)(&ldsF[0]);
    const int g = tid >> 4;              // row group 0..15
    const int l = tid & 15;
#pragma unroll
    for (int rr = 0; rr < 64; rr += 16) {
      const int k = rr + g;              // slot 0..63
      const int q = k >> 2, j = k & 3;   // recompute gather index inline
      const size_t qb = ((size_t)(b * S_ + s) * Q_ + q0 + q) * 2;
      const float py = qp[qb + 0];
      const float px = qp[qb + 1];
      const int y0 = (int)floorf(py), x0 = (int)floorf(px);
      const int yy = min(y0 + (j >> 1), SIZE_ - 1);
      const int xx = min(x0 + (j & 1),  SIZE_ - 1);
      const int n  = yy * SIZE_ + xx;
      const unsigned long long gbase =
          (unsigned long long)(uintptr_t)(features + ((size_t)b * N_ + n) * C_ +
                                          (size_t)half * CH_);
#pragma unroll
      for (int cc = 0; cc < 2; ++cc) {
        const int c4 = cc * 16 + l;      // 16B chunk index 0..31 (128 ch)
        const unsigned long long ga = gbase + (unsigned)(c4 * 16);
        const unsigned la = lds_base + (unsigned)(k * (STR_ * 4) + c4 * 16);
        asm volatile("global_load_async_to_lds_b128 %0, %1, off"
                     :: "v"(la), "v"(ga) : "memory");
      }
    }
    asm volatile("s_wait_asynccnt 0x0" ::: "memory");
  }
  __syncthreads();

  // ---- 16 chained exact-f32 WMMAs: D += A_t x B_t (K=4 per query) ----
  const int c_local = wave * 16 + row;   // this lane's channel within the half
  v8f acc = {};
#pragma unroll
  for (int t = 0; t < 16; ++t) {
    // A 16x4 f32 layout: lanes 0-15 hold K0,K1; lanes 16-31 hold K2,K3.
    v2f a;
    a.x = (row == t) ? wA : 0.0f;
    a.y = (row == t) ? wB : 0.0f;
    // B 4x16 f32 layout mirrors A's K split: lane half picks K pair.
    const float* pb = &ldsF[(4 * t + hi * 2) * STR_ + c_local];
    v2f bv;
    bv.x = pb[0];
    bv.y = pb[STR_];
    acc = __builtin_amdgcn_wmma_f32_16x16x4_f32(
        false, a, false, bv, (short)0, acc, false, false);
  }

  // ---- Store 16x16 f32 tile: lane half selects rows M = v (+8) ----
  float* __restrict__ dst =
      out + ((size_t)(b * S_ + s) * Q_ + q0) * C_ + half * CH_ + c_local;
#pragma unroll
  for (int v = 0; v < 8; ++v) {
    const int m = v + hi * 8;
    dst[(size_t)m * C_] = acc[v];
  }
}

extern "C" void kernel_launch(void* const* d_in, const int* in_sizes, int n_in,
                              void* d_out, int out_size, void* d_ws, size_t ws_size,
                              hipStream_t stream) {
  const float* features = (const float*)d_in[0];
  const float* qpts     = (const float*)d_in[1];
  float*       out      = (float*)d_out;
  (void)in_sizes; (void)n_in; (void)out_size; (void)d_ws; (void)ws_size;

  dim3 grid(B_ * S_ * (Q_ / 16) * 2);   // 1024: query tile x channel half
  dim3 block(256);                      // 8 wave32 waves
  hipLaunchKernelGGL(sqe_wmma_f32_kernel, grid, block, 0, stream,
                     features, qpts, out);
}